// InterpolationBlock2D_Quad_26010321944825
// MI455X (gfx1250) — compile-verified
//
#include <hip/hip_runtime.h>
#include <stdint.h>

// Interpolation gather-dot:
//   out[d][p] = sum_k sf[p][k] * nodal[d][ conn[cell_id[p]][k] - 1 ],  k=0..5, d=0..1
//
// Memory-bound (0.6 FLOP/byte): 288 MB mandatory streaming -> ~12.4 us floor at
// 23.3 TB/s. Gather tables (conn 24MB + nodal 4MB) are L2-resident (192MB L2).
// CDNA5 path: double-buffered async global->LDS staging (ASYNCcnt) of BOTH
// coalesced input streams (shape_functions 192MB, cell_id 32MB), overlapping
// them with the random L2 gathers and shortening the dependent-load chain
// at each tile from 3 memory hops to 2.

#define TILE   256   // points per block per pipeline stage
#define BLOCK  256   // 8 wave32s

__global__ __launch_bounds__(BLOCK)
void interp2d_quad_kernel(const float* __restrict__ nodal,    // (2, n_nodes)
                          const float* __restrict__ sf,       // (n_pts, 6)
                          const int*   __restrict__ conn,     // (n_cells, 6)
                          const int*   __restrict__ cell_id,  // (n_pts,)
                          float*       __restrict__ out,      // (2, n_pts)
                          int n_nodes, int n_pts, int n_tiles)
{
    __shared__ float smem[2 * TILE * 6];  // 12 KB: sf double buffer (24 B/pt)
    __shared__ int   scid[2 * TILE];      //  2 KB: cell_id double buffer

    const int tid = threadIdx.x;
    const int stride = gridDim.x;
    // Low 32 bits of a generic pointer into LDS == LDS byte offset
    // (async_to_lds dsaddr is LDS_BASE-relative, same convention).
    const unsigned lds_sf  = (unsigned)(uintptr_t)(&smem[0]);
    const unsigned lds_cid = (unsigned)(uintptr_t)(&scid[0]);

    const float* __restrict__ nv0 = nodal;
    const float* __restrict__ nv1 = nodal + n_nodes;

    int t = (int)blockIdx.x;
    if (t < n_tiles) {
        // ---- prologue: stage tile t into buffer 0 (4 async ops/lane)
        {
            unsigned ls = lds_sf  + (unsigned)tid * 24u;
            unsigned lc = lds_cid + (unsigned)tid * 4u;
            unsigned gs = (unsigned)(t * TILE + tid) * 24u;
            unsigned gc = (unsigned)(t * TILE + tid) * 4u;
            asm volatile(
                "global_load_async_to_lds_b64 %0, %2, %4 offset:0\n\t"
                "global_load_async_to_lds_b64 %0, %2, %4 offset:8\n\t"
                "global_load_async_to_lds_b64 %0, %2, %4 offset:16\n\t"
                "global_load_async_to_lds_b32 %1, %3, %5 offset:0"
                :: "v"(ls), "v"(lc), "v"(gs), "v"(gc), "s"(sf), "s"(cell_id)
                : "memory");
        }
        int buf = 0;
        for (; t < n_tiles; t += stride) {
            const int tn = t + stride;
            if (tn < n_tiles) {
                // Stage next tile into the other buffer, then wait for the
                // current tile's 4 async ops. Async loads complete in order,
                // so asynccnt<=4 leaves only the 4 just-issued ops pending.
                unsigned ls = lds_sf  + (unsigned)(buf ^ 1) * (TILE * 24u)
                                      + (unsigned)tid * 24u;
                unsigned lc = lds_cid + (unsigned)(buf ^ 1) * (TILE * 4u)
                                      + (unsigned)tid * 4u;
                unsigned gs = (unsigned)(tn * TILE + tid) * 24u;
                unsigned gc = (unsigned)(tn * TILE + tid) * 4u;
                asm volatile(
                    "global_load_async_to_lds_b64 %0, %2, %4 offset:0\n\t"
                    "global_load_async_to_lds_b64 %0, %2, %4 offset:8\n\t"
                    "global_load_async_to_lds_b64 %0, %2, %4 offset:16\n\t"
                    "global_load_async_to_lds_b32 %1, %3, %5 offset:0"
                    :: "v"(ls), "v"(lc), "v"(gs), "v"(gc), "s"(sf), "s"(cell_id)
                    : "memory");
                asm volatile("s_wait_asynccnt 0x4" ::: "memory");
            } else {
                asm volatile("s_wait_asynccnt 0x0" ::: "memory");
            }

            // ---- consume current tile (cell_id already resident in LDS)
            const int p   = t * TILE + tid;
            const int cid = scid[buf * TILE + tid];            // bank-conflict-free
            const int2* c2 = reinterpret_cast<const int2*>(conn + (size_t)cid * 6);
            const int2 ca = c2[0], cb = c2[1], cc = c2[2];     // 8B-aligned, L2 hits

            const float* s6 = &smem[buf * (TILE * 6) + tid * 6]; // conflict-free
            const float s0 = s6[0], s1 = s6[1], s2 = s6[2];
            const float s3 = s6[3], s4 = s6[4], s5 = s6[5];

            const int n0 = ca.x - 1, n1 = ca.y - 1, n2 = cb.x - 1;
            const int n3 = cb.y - 1, n4 = cc.x - 1, n5 = cc.y - 1;

            float a0 = s0 * nv0[n0] + s1 * nv0[n1] + s2 * nv0[n2]
                     + s3 * nv0[n3] + s4 * nv0[n4] + s5 * nv0[n5];
            float a1 = s0 * nv1[n0] + s1 * nv1[n1] + s2 * nv1[n2]
                     + s3 * nv1[n3] + s4 * nv1[n4] + s5 * nv1[n5];

            out[p]         = a0;   // dim 0, coalesced
            out[n_pts + p] = a1;   // dim 1, coalesced
            buf ^= 1;
        }
    }

    // ---- tail: points beyond the last full tile (none for 8M/256, kept for safety)
    const int full = n_tiles * TILE;
    for (int p = full + (int)blockIdx.x * BLOCK + tid; p < n_pts;
         p += stride * BLOCK) {
        const int cid = cell_id[p];
        const int2* c2 = reinterpret_cast<const int2*>(conn + (size_t)cid * 6);
        const int2 ca = c2[0], cb = c2[1], cc = c2[2];
        const float* sp = sf + (size_t)p * 6;
        const float s0 = sp[0], s1 = sp[1], s2 = sp[2];
        const float s3 = sp[3], s4 = sp[4], s5 = sp[5];
        const int n0 = ca.x - 1, n1 = ca.y - 1, n2 = cb.x - 1;
        const int n3 = cb.y - 1, n4 = cc.x - 1, n5 = cc.y - 1;
        out[p]         = s0 * nv0[n0] + s1 * nv0[n1] + s2 * nv0[n2]
                       + s3 * nv0[n3] + s4 * nv0[n4] + s5 * nv0[n5];
        out[n_pts + p] = s0 * nv1[n0] + s1 * nv1[n1] + s2 * nv1[n2]
                       + s3 * nv1[n3] + s4 * nv1[n4] + s5 * nv1[n5];
    }
}

extern "C" void kernel_launch(void* const* d_in, const int* in_sizes, int n_in,
                              void* d_out, int out_size, void* d_ws, size_t ws_size,
                              hipStream_t stream) {
    (void)n_in; (void)out_size; (void)d_ws; (void)ws_size;

    const float* nodal   = (const float*)d_in[0];  // (2, n_nodes)
    const float* sfun    = (const float*)d_in[1];  // (n_pts, 6)
    const int*   conn    = (const int*)  d_in[2];  // (n_cells, 6)
    const int*   cell_id = (const int*)  d_in[3];  // (n_pts,)
    float*       out     = (float*)d_out;          // (2, n_pts)

    const int n_nodes = in_sizes[0] / 2;
    const int n_pts   = in_sizes[3];
    const int n_tiles = n_pts / TILE;

    int blocks = 2048;                 // ~15 tiles/block -> pipeline stays warm
    if (n_tiles > 0 && n_tiles < blocks) blocks = n_tiles;
    if (blocks < 1) blocks = 1;

    interp2d_quad_kernel<<<blocks, BLOCK, 0, stream>>>(
        nodal, sfun, conn, cell_id, out, n_nodes, n_pts, n_tiles);
}